// PathAttentionScore_80633716015120
// MI455X (gfx1250) — compile-verified
//
#include <hip/hip_runtime.h>

typedef __attribute__((ext_vector_type(2))) float v2f;
typedef __attribute__((ext_vector_type(8))) float v8f;

#define N_NODES_C 100000
#define HIDDEN_C  128
#define MAXLEN_C  3

// ---------------------------------------------------------------------------
// Phase 1: proj[n][l] = dot(node_feature[n,:], W[l,0,:])  via WMMA f32 16x16x4
// One wave32 computes one 16-row tile: C(16x16) = A(16x128) x B(128x16),
// where B columns 0..2 are W[l], columns 3..15 are zero.
// B load is branch-free: clamp column to a valid W row, zero via multiply.
// ---------------------------------------------------------------------------
__global__ __launch_bounds__(256) void proj_wmma_kernel(
    const float* __restrict__ nf, const float* __restrict__ W,
    float* __restrict__ proj, int n_tiles)
{
    const int lane = threadIdx.x & 31;
    const int wave = threadIdx.x >> 5;
    const int tile = blockIdx.x * 8 + wave;
    if (tile >= n_tiles) return;          // wave-uniform: EXEC all-ones inside

    const int  m  = lane & 15;            // A row / B,C column index
    const int  hi = lane >> 4;            // lane half selects K pair
    const long long row = (long long)tile * 16 + m;
    const float* __restrict__ arow = nf + row * HIDDEN_C;

    const bool  bvalid = (m < MAXLEN_C);
    const float bscale = bvalid ? 1.0f : 0.0f;           // loop-invariant mask
    const float* __restrict__ bcol = W + (bvalid ? m : 0) * HIDDEN_C;

    v8f c = {};
    #pragma unroll
    for (int kk = 0; kk < HIDDEN_C / 4; ++kk) {
        const int k0 = kk * 4 + hi * 2;
        // A 16x4 f32: lanes 0-15 -> K = {0,1}, lanes 16-31 -> K = {2,3}
        v2f a = *(const v2f*)(arow + k0);
        // B 4x16 f32: same K striping, N = lane&15 (cols >= 3 forced to 0)
        v2f braw = *(const v2f*)(bcol + k0);
        v2f b;
        b.x = braw.x * bscale;
        b.y = braw.y * bscale;
        c = __builtin_amdgcn_wmma_f32_16x16x4_f32(
                /*neg_a=*/false, a, /*neg_b=*/false, b,
                /*c_mod=*/(short)0, c, /*reuse_a=*/false, /*reuse_b=*/false);
    }

    // C layout: VGPR r -> M = r + 8*hi, N = lane&15. Keep only N < 3.
    if (m < MAXLEN_C) {
        const long long base = (long long)tile * 16 + hi * 8;
        #pragma unroll
        for (int r = 0; r < 8; ++r)
            proj[(base + r) * MAXLEN_C + m] = c[r];
    }
}

// ---------------------------------------------------------------------------
// Phase 2: per-path masked gather + mean.
// out[i] = sum_{l: paths[i,l] >= 0} proj[paths[i,l]][l] / max(1, #valid)
// ---------------------------------------------------------------------------
__global__ __launch_bounds__(256) void path_gather_kernel(
    const int* __restrict__ paths, const float* __restrict__ proj,
    float* __restrict__ out, int n_paths)
{
    const int i = blockIdx.x * blockDim.x + threadIdx.x;
    if (i >= n_paths) return;

    const int* __restrict__ p = paths + (long long)i * MAXLEN_C;
    const int i0 = p[0];
    const int i1 = p[1];
    const int i2 = p[2];

    float s = 0.0f;
    int cnt = 0;
    if (i0 >= 0) { s += proj[(long long)i0 * MAXLEN_C + 0]; ++cnt; }
    if (i1 >= 0) { s += proj[(long long)i1 * MAXLEN_C + 1]; ++cnt; }
    if (i2 >= 0) { s += proj[(long long)i2 * MAXLEN_C + 2]; ++cnt; }

    out[i] = s / (float)(cnt > 0 ? cnt : 1);
}

extern "C" void kernel_launch(void* const* d_in, const int* in_sizes, int n_in,
                              void* d_out, int out_size, void* d_ws, size_t ws_size,
                              hipStream_t stream) {
    const int*   paths = (const int*)d_in[0];     // (10M, 3) int
    const float* nf    = (const float*)d_in[1];   // (100000, 128) f32
    const float* W     = (const float*)d_in[2];   // (3, 1, 128) f32
    float*       out   = (float*)d_out;           // (10M,) f32
    float*       proj  = (float*)d_ws;            // needs 100000*3*4 = 1.2 MB

    const int n_tiles = N_NODES_C / 16;           // 6250
    proj_wmma_kernel<<<(n_tiles + 7) / 8, 256, 0, stream>>>(nf, W, proj, n_tiles);

    const int n_paths = out_size;                 // 10,000,000 (HEAD == 1)
    path_gather_kernel<<<(n_paths + 255) / 256, 256, 0, stream>>>(paths, proj, out, n_paths);
}